// Clustering_80461917323374
// MI455X (gfx1250) — compile-verified
//
#include <hip/hip_runtime.h>
#include <math.h>

#define B_ 32
#define H_ 8
#define L_ 256
#define DK_ 64
#define C_ 3
#define DM_ 512
#define U_ 30
#define BATCH_STRIDE (H_ * L_ * DK_) /* 131072 floats per batch of Q/K/V */

typedef __attribute__((ext_vector_type(2))) float v2f;
typedef __attribute__((ext_vector_type(8))) float v8f;
typedef int gvec4i __attribute__((vector_size(16)));  // matches clang's V4i

#if __has_builtin(__builtin_amdgcn_global_load_async_to_lds_b128) && \
    __has_builtin(__builtin_amdgcn_s_wait_asynccnt)
#define USE_ASYNC_LDS 1
#endif

__device__ __forceinline__ float gelu_f(float x) {
  // exact (erf) GELU, matching jax.nn.gelu(approximate=False)
  return 0.5f * x * (1.0f + erff(x * 0.70710678118654752f));
}

__device__ __forceinline__ float wave_sum(float v) {
#pragma unroll
  for (int m = 16; m > 0; m >>= 1) v += __shfl_xor(v, m, 32);
  return v;
}

__device__ __forceinline__ float wave_max(float v) {
#pragma unroll
  for (int m = 16; m > 0; m >>= 1) v = fmaxf(v, __shfl_xor(v, m, 32));
  return v;
}

// ---- cooperative 16B-per-lane copy: global -> LDS (async path if available)
__device__ __forceinline__ void copy16(const float* __restrict__ g, float* s) {
#ifdef USE_ASYNC_LDS
  __builtin_amdgcn_global_load_async_to_lds_b128(
      (__attribute__((address_space(1))) gvec4i*)g,
      (__attribute__((address_space(3))) gvec4i*)s, 0, 0);
#else
  *(float4*)s = *(const float4*)g;
#endif
}

__device__ __forceinline__ void async_join() {
#ifdef USE_ASYNC_LDS
  __builtin_amdgcn_s_wait_asynccnt(0);
#endif
}

// Stage slab (32 rows x 512, natural batch order) + 512x3 weights into LDS.
__device__ __forceinline__ void stage_phase(const float* __restrict__ X, int l,
                                            const float* __restrict__ W,
                                            float* sSlab, float* sW, int tid) {
#pragma unroll
  for (int i = 0; i < (B_ * DM_ / 4) / 1024; ++i) {  // 4 float4s per thread
    const int idx = tid + i * 1024;                  // float4 index, 0..4095
    const int bp = idx >> 7;                         // 128 float4 per row
    const int t4 = (idx & 127) * 4;
    copy16(X + (size_t)bp * BATCH_STRIDE + (size_t)l * DM_ + t4,
           sSlab + bp * DM_ + t4);
  }
  if (tid < (DM_ * C_) / 4) copy16(W + tid * 4, sW + tid * 4);
}

// Fetch Ku[u][m], Ku[u][m+1] from the natural-order slab:
//   Ku[u][m] = slab[(b-29 + s%30)*512 + s/30],  s = 512u + m  (0 if batch < 0)
__device__ __forceinline__ v2f slab_fetch2(const float* sSlab, int s,
                                           int bm29) {
  const int t = s / U_;
  const int u2 = s - t * U_;
  const bool wrap = (u2 == U_ - 1);
  const int t1 = wrap ? t + 1 : t;
  const int u2b = wrap ? 0 : u2 + 1;
  const int bp0 = bm29 + u2;
  const int bp1 = bm29 + u2b;
  const float v0 = sSlab[(bp0 >= 0 ? bp0 : 0) * DM_ + t];
  const float v1 = sSlab[(bp1 >= 0 ? bp1 : 0) * DM_ + t1];
  v2f r;
  r.x = (bp0 >= 0) ? v0 : 0.f;
  r.y = (bp1 >= 0) ? v1 : 0.f;
  return r;
}

// D[u][c] += Ku[u][k..k+3] * W[k..k+3][c] via fp32 WMMA (16x16x4).
// M = u (rows 0..29 live), N = c (cols 0..2 live), K = 512.
__device__ __forceinline__ void proj_wmma(const float* sSlab, const float* sW,
                                          int lane, int bm29, v8f& acc0,
                                          v8f& acc1) {
  const int nlo = lane & 15;
  const int hi = lane >> 4;
  const int u1 = nlo + 16;
  const bool live1 = (u1 < U_);
  const int ccol = (nlo < C_) ? nlo : 0;
  const bool bcol = (nlo < C_);
#pragma unroll 2
  for (int k = 0; k < DM_; k += 4) {
    const int kk = k + 2 * hi;  // lanes 0-15: K=k,k+1 ; lanes 16-31: K=k+2,k+3
    v2f a0 = slab_fetch2(sSlab, nlo * DM_ + kk, bm29);
    v2f a1 = slab_fetch2(sSlab, live1 ? (u1 * DM_ + kk) : 0, bm29);
    if (!live1) { a1.x = 0.f; a1.y = 0.f; }
    v2f bf;
    bf.x = sW[kk * C_ + ccol];
    bf.y = sW[(kk + 1) * C_ + ccol];
    if (!bcol) { bf.x = 0.f; bf.y = 0.f; }
    acc0 = __builtin_amdgcn_wmma_f32_16x16x4_f32(false, a0, false, bf, (short)0,
                                                 acc0, false, false);
    acc1 = __builtin_amdgcn_wmma_f32_16x16x4_f32(false, a1, false, bf, (short)0,
                                                 acc1, false, false);
  }
}

extern "C" __global__ __launch_bounds__(1024) void cluster_main(
    const float* __restrict__ Q, const float* __restrict__ K,
    const float* __restrict__ V, const float* __restrict__ Wq,
    const float* __restrict__ bq, const float* __restrict__ Wk,
    const float* __restrict__ bk, const float* __restrict__ Wv,
    const float* __restrict__ bv, const float* __restrict__ Wb,
    const float* __restrict__ bb, float* __restrict__ out,
    float* __restrict__ ws_mu, float* __restrict__ ws_lse) {
  __shared__ float sSlab[B_ * DM_];    // 64 KB: K rows then V rows (reused)
  __shared__ float sW[DM_ * C_];       // 6 KB: Wk then Wv (reused)
  __shared__ float sCk[B_ * 32 * C_];  // 12 KB
  __shared__ float sCv[B_ * 32 * C_];  // 12 KB

  const int l = blockIdx.x;
  const int tid = threadIdx.x;
  const int lane = tid & 31;
  const int b = tid >> 5;  // wave id == batch index
  const int bm29 = b - (U_ - 1);
  const int nlo = lane & 15;
  const int hi = lane >> 4;
  const size_t rowoff = ((size_t)b * L_ + l) * DM_;  // Q / out flat row

  // ---------------- stage K slab + Wk; overlap with cq ----------------
  stage_phase(K, l, Wk, sSlab, sW, tid);

  float p0 = 0.f, p1 = 0.f, p2 = 0.f;
#pragma unroll
  for (int i = 0; i < DM_ / 32; ++i) {
    const int m = lane + 32 * i;
    const float q = Q[rowoff + m];
    p0 += q * Wq[m * C_ + 0];
    p1 += q * Wq[m * C_ + 1];
    p2 += q * Wq[m * C_ + 2];
  }
  p0 = wave_sum(p0);
  p1 = wave_sum(p1);
  p2 = wave_sum(p2);
  const float cq0 = gelu_f(p0 + bq[0]);
  const float cq1 = gelu_f(p1 + bq[1]);
  const float cq2 = gelu_f(p2 + bq[2]);
  if (lane == 0) ws_mu[b * L_ + l] = (cq0 + cq1 + cq2) * (1.0f / 3.0f);

  async_join();
  __syncthreads();

  // ---------------- ck = gelu(Ku @ Wk + bk) via WMMA ----------------
  {
    v8f acc0 = {0.f, 0.f, 0.f, 0.f, 0.f, 0.f, 0.f, 0.f};
    v8f acc1 = {0.f, 0.f, 0.f, 0.f, 0.f, 0.f, 0.f, 0.f};
    proj_wmma(sSlab, sW, lane, bm29, acc0, acc1);
    if (nlo < C_) {
      const float bias = bk[nlo];
      float* dst = sCk + b * (32 * C_);
#pragma unroll
      for (int r = 0; r < 8; ++r) {  // row = r + 8*hi (tile0), +16 (tile1)
        dst[(r + 8 * hi) * C_ + nlo] = gelu_f(acc0[r] + bias);
        dst[(16 + r + 8 * hi) * C_ + nlo] = gelu_f(acc1[r] + bias);
      }
    }
  }
  __syncthreads();  // all waves done reading K slab / Wk

  // ---------------- stage V slab + Wv ----------------
  stage_phase(V, l, Wv, sSlab, sW, tid);
  async_join();
  __syncthreads();

  // ---------------- cv = gelu(Vu @ Wv + bv) via WMMA ----------------
  {
    v8f acc0 = {0.f, 0.f, 0.f, 0.f, 0.f, 0.f, 0.f, 0.f};
    v8f acc1 = {0.f, 0.f, 0.f, 0.f, 0.f, 0.f, 0.f, 0.f};
    proj_wmma(sSlab, sW, lane, bm29, acc0, acc1);
    if (nlo < C_) {
      const float bias = bv[nlo];
      float* dst = sCv + b * (32 * C_);
#pragma unroll
      for (int r = 0; r < 8; ++r) {
        dst[(r + 8 * hi) * C_ + nlo] = gelu_f(acc0[r] + bias);
        dst[(16 + r + 8 * hi) * C_ + nlo] = gelu_f(acc1[r] + bias);
      }
    }
  }
  __syncthreads();  // sCk/sCv visible to all lanes

  // ---------------- softmax over u, ctx, loss pieces ----------------
  const float* ckb = sCk + b * (32 * C_);
  const float* cvb = sCv + b * (32 * C_);
  float sc;
  if (lane < U_)
    sc = (cq0 * ckb[lane * C_ + 0] + cq1 * ckb[lane * C_ + 1] +
          cq2 * ckb[lane * C_ + 2]) *
         0.04419417382415922f;  // 1/sqrt(512)
  else
    sc = -3.0e38f;
  const float mx = wave_max(sc);
  const float e = (lane < U_) ? expf(sc - mx) : 0.f;
  const float den = wave_sum(e);
  const float attn = e / den;

  const float t = (lane < U_) ? expf(attn) : 0.f;  // logsumexp of attn probs
  const float lse = logf(wave_sum(t));
  if (lane == 0) ws_lse[b * L_ + l] = lse;

  float c0 = (lane < U_) ? attn * cvb[lane * C_ + 0] : 0.f;
  float c1 = (lane < U_) ? attn * cvb[lane * C_ + 1] : 0.f;
  float c2 = (lane < U_) ? attn * cvb[lane * C_ + 2] : 0.f;
  c0 = wave_sum(c0);
  c1 = wave_sum(c1);
  c2 = wave_sum(c2);

  // ---------------- out = gelu(ctx @ Wb + bb) ----------------
#pragma unroll
  for (int i = 0; i < DM_ / 32; ++i) {
    const int m = lane + 32 * i;
    const float v =
        bb[m] + c0 * Wb[m] + c1 * Wb[DM_ + m] + c2 * Wb[2 * DM_ + m];
    out[rowoff + m] = gelu_f(v);
  }
}

extern "C" __global__ __launch_bounds__(256) void loss_kernel(
    const float* __restrict__ mu, const float* __restrict__ lse,
    float* __restrict__ out_loss) {
  __shared__ float red[256];
  const int l = threadIdx.x;

  // sum of lse over all (b,l)
  float s = 0.f;
  for (int b = 0; b < B_; ++b) s += lse[b * L_ + l];
  red[l] = s;
  __syncthreads();
  for (int off = 128; off > 0; off >>= 1) {
    if (l < off) red[l] += red[l + off];
    __syncthreads();
  }
  const float lse_sum = red[0];
  __syncthreads();

  // ce = -(1/B) * sum_b sum_l mu * log_softmax_l(mu)
  float ce_total = 0.f;
  for (int b = 0; b < B_; ++b) {
    const float m = mu[b * L_ + l];
    red[l] = m;
    __syncthreads();
    for (int off = 128; off > 0; off >>= 1) {
      if (l < off) red[l] = fmaxf(red[l], red[l + off]);
      __syncthreads();
    }
    const float mxv = red[0];
    __syncthreads();
    red[l] = expf(m - mxv);
    __syncthreads();
    for (int off = 128; off > 0; off >>= 1) {
      if (l < off) red[l] += red[l + off];
      __syncthreads();
    }
    const float logden = logf(red[0]) + mxv;
    __syncthreads();
    red[l] = m * (m - logden);
    __syncthreads();
    for (int off = 128; off > 0; off >>= 1) {
      if (l < off) red[l] += red[l + off];
      __syncthreads();
    }
    if (l == 0) ce_total += red[0];
    __syncthreads();
  }

  if (l == 0) {
    const float ce = -(ce_total / (float)B_);
    out_loss[0] = -(lse_sum / (float)(B_ * L_)) + 0.001f * ce;
  }
}

extern "C" void kernel_launch(void* const* d_in, const int* in_sizes, int n_in,
                              void* d_out, int out_size, void* d_ws,
                              size_t ws_size, hipStream_t stream) {
  const float* Q = (const float*)d_in[0];
  const float* K = (const float*)d_in[1];
  const float* V = (const float*)d_in[2];
  const float* Wq = (const float*)d_in[3];
  const float* bq = (const float*)d_in[4];
  const float* Wk = (const float*)d_in[5];
  const float* bk = (const float*)d_in[6];
  const float* Wv = (const float*)d_in[7];
  const float* bv = (const float*)d_in[8];
  const float* Wb = (const float*)d_in[9];
  const float* bb = (const float*)d_in[10];

  float* out = (float*)d_out;
  float* ws_mu = (float*)d_ws;      // B*L floats
  float* ws_lse = ws_mu + B_ * L_;  // B*L floats

  cluster_main<<<dim3(L_), 1024, 0, stream>>>(Q, K, V, Wq, bq, Wk, bk, Wv, bv,
                                              Wb, bb, out, ws_mu, ws_lse);
  loss_kernel<<<1, 256, 0, stream>>>(ws_mu, ws_lse,
                                     out + (size_t)B_ * H_ * L_ * DK_);
}